// ONNXExportWrapper_69389491634431
// MI455X (gfx1250) — compile-verified
//
#include <hip/hip_runtime.h>
#include <hip/hip_bf16.h>

// ---------------------------------------------------------------------------
// Diagonal SSM scan reformulated as chunked GEMMs (SSD-style) for CDNA5 WMMA.
//   per d:  Y = Mhat @ U  +  CA @ S_prev        (16x32x16 and 16x32x64 GEMMs)
//           S = a^16 . S_prev + AB @ U          (64x32x16 GEMM)
// All GEMMs built from V_WMMA_F32_16X16X4_F32 (wave32, A/B = v2f, C/D = v8f).
// ---------------------------------------------------------------------------

typedef float v2f __attribute__((ext_vector_type(2)));
typedef float v8f __attribute__((ext_vector_type(8)));

#define T_LEN    4096
#define BATCH    32
#define D_MODEL  256
#define NSTATE   64
#define CHUNK_L  16
#define NCHUNKS  (T_LEN / CHUNK_L)
#define N_CLASSES 5

static __device__ __forceinline__ v8f wmma_f32(v2f a, v2f b, v8f c) {
    // 8-arg form: (neg_a, A, neg_b, B, c_mod, C, reuse_a, reuse_b)
    return __builtin_amdgcn_wmma_f32_16x16x4_f32(false, a, false, b, (short)0, c,
                                                 false, false);
}

static __device__ __forceinline__ float gelu_approx(float x) {
    // tanh-approx GELU written as x * sigmoid(2k(x + 0.044715 x^3)) (overflow-safe)
    float z = 0.7978845608028654f * (x + 0.044715f * x * x * x);
    return x / (1.0f + __expf(-2.0f * z));
}

// ---------------------------------------------------------------------------
// Kernel 0: transpose x[B][T] -> xt[T][B] so chunk loads are coalesced.
// ---------------------------------------------------------------------------
__global__ void transpose_x(const float* __restrict__ x, float* __restrict__ xt) {
    int idx = blockIdx.x * blockDim.x + threadIdx.x;   // = t*32 + b
    int t = idx >> 5;
    int b = idx & 31;
    xt[idx] = x[b * T_LEN + t];
}

// ---------------------------------------------------------------------------
// Kernel 1: one block per channel d; 2 waves; wave w owns batches 16w..16w+15.
// ---------------------------------------------------------------------------
__global__ void __launch_bounds__(64)
ssm_chunk_kernel(const float* __restrict__ xt,
                 const float* __restrict__ w_in,  const float* __restrict__ b_in,
                 const float* __restrict__ A_diag, const float* __restrict__ B_inp,
                 const float* __restrict__ C_out,  const float* __restrict__ D_skip,
                 float* __restrict__ pooled /* [32][512] */) {
    __shared__ float apow[NSTATE][17];   // a_n^0 .. a_n^16
    __shared__ float cs[NSTATE], bs[NSTATE];
    __shared__ float kkArr[CHUNK_L];
    __shared__ float Mhat[CHUNK_L][CHUNK_L];
    __shared__ float CAm[CHUNK_L][NSTATE];
    __shared__ float ABm[NSTATE][CHUNK_L];
    __shared__ float aL[NSTATE];
    __shared__ float sP[2][16][66];      // per-wave carry state, [b][n], padded

    const int d    = blockIdx.x;
    const int tid  = threadIdx.x;
    const int lane = tid & 31;
    const int wave = tid >> 5;
    const int half = lane >> 4;          // 0: lanes 0-15, 1: lanes 16-31
    const int lid  = lane & 15;

    const float wd = w_in[d];
    const float bd = b_in[d];
    const float Dd = D_skip[d];

    // ---- stage 1: per-state constants + powers of a ----
    if (tid < NSTATE) {
        float a = A_diag[d * NSTATE + tid];
        cs[tid] = C_out[d * NSTATE + tid];
        bs[tid] = B_inp[d * NSTATE + tid];
        float p = 1.0f;
        #pragma unroll
        for (int q = 0; q < 17; ++q) { apow[tid][q] = p; p *= a; }
        aL[tid] = apow[tid][16];
    }
    for (int e = tid; e < 2 * 16 * 66; e += 64) ((float*)sP)[e] = 0.0f;
    __syncthreads();

    // ---- stage 2: kk (Toeplitz kernel), CA, AB tables ----
    if (tid < CHUNK_L) {
        float s = 0.0f;
        for (int n = 0; n < NSTATE; ++n) s += cs[n] * bs[n] * apow[n][tid];
        kkArr[tid] = s;
    }
    for (int e = tid; e < CHUNK_L * NSTATE; e += 64) {
        int i = e >> 6, n = e & 63;
        CAm[i][n] = cs[n] * apow[n][i + 1];
    }
    for (int e = tid; e < NSTATE * CHUNK_L; e += 64) {
        int n = e >> 4, j = e & 15;
        ABm[n][j] = bs[n] * apow[n][15 - j];
    }
    __syncthreads();

    for (int e = tid; e < CHUNK_L * CHUNK_L; e += 64) {
        int i = e >> 4, j = e & 15;
        Mhat[i][j] = (j > i) ? 0.0f : (kkArr[i - j] + ((i == j) ? Dd : 0.0f));
    }
    __syncthreads();

    // ---- stage 3: build constant WMMA A-fragments in registers ----
    v2f mhatA[4];                 // Mhat [16x16], 4 K-steps
    v2f caA[16];                  // CA   [16x64], 16 K-steps
    v2f abA[4][4];                // AB   [64x16], 4 M-tiles x 4 K-steps
    v8f aLv[4];                   // a^16 arranged to D-layout of state tiles
    v8f sacc[4];                  // state accumulators (D-layout)

    #pragma unroll
    for (int k = 0; k < 4; ++k) {
        int c0 = 4 * k + 2 * half;
        mhatA[k][0] = Mhat[lid][c0];
        mhatA[k][1] = Mhat[lid][c0 + 1];
    }
    #pragma unroll
    for (int k = 0; k < 16; ++k) {
        int c0 = 4 * k + 2 * half;
        caA[k][0] = CAm[lid][c0];
        caA[k][1] = CAm[lid][c0 + 1];
    }
    #pragma unroll
    for (int m = 0; m < 4; ++m) {
        #pragma unroll
        for (int k = 0; k < 4; ++k) {
            int c0 = 4 * k + 2 * half;
            abA[m][k][0] = ABm[16 * m + lid][c0];
            abA[m][k][1] = ABm[16 * m + lid][c0 + 1];
        }
        #pragma unroll
        for (int v = 0; v < 8; ++v) {
            aLv[m][v]  = aL[16 * m + v + 8 * half];
            sacc[m][v] = 0.0f;
        }
    }

    // ---- stage 4: sequential chunk loop (carry dependency), WMMA-dense ----
    const int bglob = 16 * wave + lid;           // this lane's batch column
    float* sPm = &sP[wave][0][0];
    float avg_acc = 0.0f;
    float max_acc = -3.402823466e38f;

    for (int ch = 0; ch < NCHUNKS; ++ch) {
        const int t0 = ch * CHUNK_L;
        if (ch + 1 < NCHUNKS)
            __builtin_prefetch(&xt[(t0 + CHUNK_L) * BATCH + bglob], 0, 0);

        // U fragments (B-layout: K = local time j, N = batch)
        v2f uf[4];
        #pragma unroll
        for (int k = 0; k < 4; ++k) {
            int j = 4 * k + 2 * half;
            float x0 = xt[(t0 + j) * BATCH + bglob];
            float x1 = xt[(t0 + j + 1) * BATCH + bglob];
            uf[k][0] = fmaf(x0, wd, bd);
            uf[k][1] = fmaf(x1, wd, bd);
        }

        // Y = Mhat @ U
        v8f y = {0.f, 0.f, 0.f, 0.f, 0.f, 0.f, 0.f, 0.f};
        #pragma unroll
        for (int k = 0; k < 4; ++k) y = wmma_f32(mhatA[k], uf[k], y);

        // Y += CA @ S_prev   (B-fragments of old state read from LDS)
        #pragma unroll
        for (int k = 0; k < 16; ++k) {
            int n0 = 4 * k + 2 * half;
            v2f bf;
            bf[0] = sPm[lid * 66 + n0];
            bf[1] = sPm[lid * 66 + n0 + 1];
            y = wmma_f32(caA[k], bf, y);
        }

        // S = a^16 . S_prev + AB @ U
        #pragma unroll
        for (int m = 0; m < 4; ++m) {
            v8f c = aLv[m] * sacc[m];
            #pragma unroll
            for (int k = 0; k < 4; ++k) c = wmma_f32(abA[m][k], uf[k], c);
            sacc[m] = c;
        }

        // publish new state for next chunk's B-fragments (per-wave region,
        // LDS ops from one wave are kept in order -> no barrier needed)
        #pragma unroll
        for (int m = 0; m < 4; ++m)
            #pragma unroll
            for (int v = 0; v < 8; ++v)
                sPm[lid * 66 + 16 * m + v + 8 * half] = sacc[m][v];

        // GELU + streaming avg/max pool (this lane holds 8 time rows)
        #pragma unroll
        for (int v = 0; v < 8; ++v) {
            float h = gelu_approx(y[v]);
            avg_acc += h;
            max_acc = fmaxf(max_acc, h);
        }
    }

    // batch b's time axis is split across lane b and lane b+16 -> combine
    float avg_o = __shfl_xor(avg_acc, 16, 32);
    float max_o = __shfl_xor(max_acc, 16, 32);
    avg_acc += avg_o;
    max_acc = fmaxf(max_acc, max_o);
    if (half == 0) {
        pooled[bglob * (2 * D_MODEL) + d]            = avg_acc * (1.0f / T_LEN);
        pooled[bglob * (2 * D_MODEL) + D_MODEL + d]  = max_acc;
    }
}

// ---------------------------------------------------------------------------
// Kernel 2: head GEMM  out[32][5] = pooled[32][512] @ W[512][5] + b  (WMMA)
// ---------------------------------------------------------------------------
__global__ void __launch_bounds__(32)
head_kernel(const float* __restrict__ pooled, const float* __restrict__ W,
            const float* __restrict__ bh, float* __restrict__ out) {
    const int lane = threadIdx.x & 31;
    const int half = lane >> 4;
    const int lid  = lane & 15;

    v8f acc0 = {0.f, 0.f, 0.f, 0.f, 0.f, 0.f, 0.f, 0.f};
    v8f acc1 = {0.f, 0.f, 0.f, 0.f, 0.f, 0.f, 0.f, 0.f};

    for (int k0 = 0; k0 < 2 * D_MODEL; k0 += 4) {
        int kA = k0 + 2 * half;
        v2f a0, a1, bf;
        a0[0] = pooled[lid * (2 * D_MODEL) + kA];
        a0[1] = pooled[lid * (2 * D_MODEL) + kA + 1];
        a1[0] = pooled[(16 + lid) * (2 * D_MODEL) + kA];
        a1[1] = pooled[(16 + lid) * (2 * D_MODEL) + kA + 1];
        bf[0] = (lid < N_CLASSES) ? W[kA * N_CLASSES + lid] : 0.0f;
        bf[1] = (lid < N_CLASSES) ? W[(kA + 1) * N_CLASSES + lid] : 0.0f;
        acc0 = wmma_f32(a0, bf, acc0);
        acc1 = wmma_f32(a1, bf, acc1);
    }

    if (lid < N_CLASSES) {
        float bb = bh[lid];
        #pragma unroll
        for (int v = 0; v < 8; ++v) {
            int r = v + 8 * half;                       // row inside 16-tile
            out[r * N_CLASSES + lid]        = acc0[v] + bb;
            out[(16 + r) * N_CLASSES + lid] = acc1[v] + bb;
        }
    }
}

// ---------------------------------------------------------------------------
extern "C" void kernel_launch(void* const* d_in, const int* in_sizes, int n_in,
                              void* d_out, int out_size, void* d_ws, size_t ws_size,
                              hipStream_t stream) {
    const float* x      = (const float*)d_in[0];
    const float* w_in   = (const float*)d_in[1];
    const float* b_in   = (const float*)d_in[2];
    const float* A_diag = (const float*)d_in[3];
    const float* B_inp  = (const float*)d_in[4];
    const float* C_out  = (const float*)d_in[5];
    const float* D_skip = (const float*)d_in[6];
    const float* W_head = (const float*)d_in[7];
    const float* b_head = (const float*)d_in[8];
    float* out = (float*)d_out;

    float* xt     = (float*)d_ws;            // [4096][32]
    float* pooled = xt + T_LEN * BATCH;      // [32][512]

    transpose_x<<<(T_LEN * BATCH) / 256, 256, 0, stream>>>(x, xt);
    ssm_chunk_kernel<<<D_MODEL, 64, 0, stream>>>(xt, w_in, b_in, A_diag, B_inp,
                                                 C_out, D_skip, pooled);
    head_kernel<<<1, 32, 0, stream>>>(pooled, W_head, b_head, out);
}